// HeteroGNN_69544110456890
// MI455X (gfx1250) — compile-verified
//
#include <hip/hip_runtime.h>
#include <hip/hip_bf16.h>

#define H 128
#define AA 64
#define HP (H + 4)          // LDS row padding -> bank-conflict-free WMMA A reads
#define MTILE 32            // rows per block in the WMMA GEMM (100000 % 32 == 0)

typedef float v2f __attribute__((ext_vector_type(2)));
typedef float v8f __attribute__((ext_vector_type(8)));

// ---------------------------------------------------------------------------
// Weight folding:  Wd' = Wdst @ Wupd[:H],  Ws' = Wsrc @ Wupd[H:],
//                  b'  = bdst@Wupd[:H] + bsrc@Wupd[H:] + bupd
// One block per (layer, relation) pair (6 total). Tiny, off the hot path.
// ---------------------------------------------------------------------------
__global__ __launch_bounds__(256) void fold_weights(
    const float* __restrict__ Wsrc, const float* __restrict__ bsrc,
    const float* __restrict__ Wdst, const float* __restrict__ bdst,
    const float* __restrict__ Wupd, const float* __restrict__ bupd,
    float* __restrict__ Wd, float* __restrict__ Ws, float* __restrict__ bf)
{
    int lr = blockIdx.x;                       // 0..5  == l*3 + r
    const float* wd = Wdst + (long)lr * H * H;
    const float* ws = Wsrc + (long)lr * H * H;
    const float* wu = Wupd + (long)lr * 2 * H * H;   // [2H][H]
    float* od = Wd + (long)lr * H * H;
    float* os = Ws + (long)lr * H * H;
    for (int idx = threadIdx.x; idx < H * H; idx += blockDim.x) {
        int i = idx / H, j = idx % H;
        float accd = 0.f, accs = 0.f;
        for (int m = 0; m < H; ++m) {
            accd += wd[i * H + m] * wu[m * H + j];
            accs += ws[i * H + m] * wu[(H + m) * H + j];
        }
        od[idx] = accd;
        os[idx] = accs;
    }
    const float* bd = bdst + lr * H;
    const float* bs = bsrc + lr * H;
    const float* bu = bupd + lr * H;
    for (int j = threadIdx.x; j < H; j += blockDim.x) {
        float acc = bu[j];
        for (int m = 0; m < H; ++m)
            acc += bd[m] * wu[m * H + j] + bs[m] * wu[(H + m) * H + j];
        bf[lr * H + j] = acc;
    }
}

// ---------------------------------------------------------------------------
// Scatter-mean phase 1: one wave per edge, 4 floats per lane, HW f32 atomics.
// ---------------------------------------------------------------------------
__global__ __launch_bounds__(256) void scatter_add(
    const float* __restrict__ xs, const int* __restrict__ esrc,
    const int* __restrict__ edst, float* __restrict__ aggr,
    float* __restrict__ cnt, int nE)
{
    long gid = (long)blockIdx.x * blockDim.x + threadIdx.x;
    int e = (int)(gid >> 5);
    if (e >= nE) return;
    int sub = (int)(gid & 31);
    int s = esrc[e], d = edst[e];
    const float4 v = reinterpret_cast<const float4*>(xs + (long)s * H)[sub];
    float* dstp = aggr + (long)d * H + sub * 4;
    unsafeAtomicAdd(dstp + 0, v.x);
    unsafeAtomicAdd(dstp + 1, v.y);
    unsafeAtomicAdd(dstp + 2, v.z);
    unsafeAtomicAdd(dstp + 3, v.w);
    if (sub == 0) unsafeAtomicAdd(cnt + d, 1.0f);
}

__global__ __launch_bounds__(256) void scatter_div(
    float* __restrict__ aggr, const float* __restrict__ cnt, long total)
{
    long i = (long)blockIdx.x * blockDim.x + threadIdx.x;
    if (i >= total) return;
    float c = cnt[i >> 7];                  // /H
    aggr[i] /= fmaxf(c, 1.0f);
}

// ---------------------------------------------------------------------------
// Fused dual-GEMM + bias:  out[n,:] = A1[n,:]@W1 + A2[n,:]@W2 + b
// MTILE=32 rows / block; 8 waves, each owns one 16-col panel and computes TWO
// 16x16 row-tiles: B fragments loaded once per K-step are reused for both
// tiles, and acc0/acc1 interleave -> 2 independent WMMA accumulate chains.
// K=128 swept in steps of 4 with V_WMMA_F32_16X16X4_F32 (exact fp32).
// Requires nN % MTILE == 0 (holds: 100000/32 = 3125).
// ---------------------------------------------------------------------------
__global__ __launch_bounds__(256) void gemm_dual_bias(
    const float* __restrict__ A1, const float* __restrict__ W1,
    const float* __restrict__ A2, const float* __restrict__ W2,
    const float* __restrict__ bias, float* __restrict__ out)
{
    __shared__ float lds[2][MTILE][HP];     // ~33 KB
    const int tid = threadIdx.x;
    const long rowBase = (long)blockIdx.x * MTILE;

    // cooperative stage of both MTILExH A panels (2048 float4 total)
    const float* srcs[2] = {A1, A2};
    for (int i = tid; i < 2 * MTILE * (H / 4); i += 256) {
        int buf = i >> 10;                // 1024 float4 per panel
        int rem = i & 1023;
        int r   = rem >> 5;               // 32 float4 per row
        int c4  = rem & 31;
        float4 v = reinterpret_cast<const float4*>(srcs[buf] + (rowBase + r) * H)[c4];
        reinterpret_cast<float4*>(&lds[buf][r][0])[c4] = v;
    }
    __syncthreads();

    const int wave  = tid >> 5;
    const int lane  = tid & 31;
    const int tileN = wave * 16;
    const int lhalf = lane >> 4;          // 0: K={0,1}, 1: K={2,3}  (A frag)
    const int l16   = lane & 15;

    v8f acc0 = {};                        // rows rowBase +  0..15
    v8f acc1 = {};                        // rows rowBase + 16..31
    for (int kk = 0; kk < H; kk += 4) {
        const int ka = kk + lhalf * 2;
        // B 4x16 f32 frags (shared by both row tiles)
        v2f b1; b1.x = W1[(ka + 0) * H + tileN + l16];
                b1.y = W1[(ka + 1) * H + tileN + l16];
        v2f b2; b2.x = W2[(ka + 0) * H + tileN + l16];
                b2.y = W2[(ka + 1) * H + tileN + l16];
        // A 16x4 f32 frags: lanes0-15 M rows, K={kk,kk+1}; lanes16-31 K={kk+2,kk+3}
        v2f a;
        a.x = lds[0][l16][ka];       a.y = lds[0][l16][ka + 1];
        acc0 = __builtin_amdgcn_wmma_f32_16x16x4_f32(false, a, false, b1,
                                                     (short)0, acc0, false, false);
        a.x = lds[0][16 + l16][ka];  a.y = lds[0][16 + l16][ka + 1];
        acc1 = __builtin_amdgcn_wmma_f32_16x16x4_f32(false, a, false, b1,
                                                     (short)0, acc1, false, false);
        a.x = lds[1][l16][ka];       a.y = lds[1][l16][ka + 1];
        acc0 = __builtin_amdgcn_wmma_f32_16x16x4_f32(false, a, false, b2,
                                                     (short)0, acc0, false, false);
        a.x = lds[1][16 + l16][ka];  a.y = lds[1][16 + l16][ka + 1];
        acc1 = __builtin_amdgcn_wmma_f32_16x16x4_f32(false, a, false, b2,
                                                     (short)0, acc1, false, false);
    }

    const float bv = bias[tileN + l16];
    const int rowOff = lhalf * 8;         // C/D: VGPR r holds M=r (lanes<16), M=r+8
    for (int r = 0; r < 8; ++r) {
        out[(rowBase + rowOff + r) * H + tileN + l16]      = acc0[r] + bv;
        out[(rowBase + 16 + rowOff + r) * H + tileN + l16] = acc1[r] + bv;
    }
}

// ---------------------------------------------------------------------------
// Attention score sums: one wave32 per (node, stack). Row held in registers
// (float4/lane); butterfly __shfl_xor reduction per attn column.
// ---------------------------------------------------------------------------
__global__ __launch_bounds__(256) void attn_score(
    const float* __restrict__ e0, const float* __restrict__ e1,
    const float* __restrict__ W1, const float* __restrict__ b1,
    const float* __restrict__ w2, float* __restrict__ sums, int nN)
{
    long wid = ((long)blockIdx.x * blockDim.x + threadIdx.x) >> 5;
    int lane = threadIdx.x & 31;
    if (wid >= 2L * nN) return;
    int k = (wid >= nN);
    long node = wid - (long)k * nN;
    const float* x = (k ? e1 : e0) + node * H;
    float4 xv = reinterpret_cast<const float4*>(x)[lane];
    float s = 0.f;
    for (int j = 0; j < AA; ++j) {
        float p = xv.x * W1[(lane * 4 + 0) * AA + j]
                + xv.y * W1[(lane * 4 + 1) * AA + j]
                + xv.z * W1[(lane * 4 + 2) * AA + j]
                + xv.w * W1[(lane * 4 + 3) * AA + j];
        for (int off = 16; off; off >>= 1) p += __shfl_xor(p, off, 32);
        s += tanhf(p + b1[j]) * w2[j];
    }
    if (lane == 0) unsafeAtomicAdd(&sums[k], s);
}

__global__ void attn_alpha(const float* __restrict__ sums,
                           float* __restrict__ alpha, float invN)
{
    float a0 = sums[0] * invN, a1 = sums[1] * invN;
    float m  = fmaxf(a0, a1);
    float x0 = __expf(a0 - m), x1 = __expf(a1 - m);
    float d  = x0 + x1;
    alpha[0] = x0 / d;
    alpha[1] = x1 / d;
}

__global__ __launch_bounds__(256) void attn_combine(
    const float* __restrict__ e0, const float* __restrict__ e1,
    const float* __restrict__ alpha, float* __restrict__ out, long total)
{
    long i = (long)blockIdx.x * blockDim.x + threadIdx.x;
    if (i >= total) return;
    out[i] = alpha[0] * e0[i] + alpha[1] * e1[i];
}

// ---------------------------------------------------------------------------
// BatchNorm: per-column sums via coalesced strided rows + f32 atomics.
// ---------------------------------------------------------------------------
#define BN_ROWS 512
__global__ __launch_bounds__(256) void bn_stats(
    const float* __restrict__ x, float* __restrict__ gsum,
    float* __restrict__ gsq, int nN)
{
    int c    = threadIdx.x & (H - 1);
    int half = threadIdx.x >> 7;          // 0/1 -> two rows in flight
    long rend = (long)(blockIdx.x + 1) * BN_ROWS;
    if (rend > nN) rend = nN;
    float s = 0.f, q = 0.f;
    for (long r = (long)blockIdx.x * BN_ROWS + half; r < rend; r += 2) {
        float v = x[r * H + c];
        s += v;
        q += v * v;
    }
    unsafeAtomicAdd(&gsum[c], s);
    unsafeAtomicAdd(&gsq[c], q);
}

__global__ void bn_finalize(const float* __restrict__ gsum,
                            const float* __restrict__ gsq,
                            float* __restrict__ mu, float* __restrict__ rstd,
                            float invN)
{
    int c = threadIdx.x;
    float m = gsum[c] * invN;
    float v = gsq[c] * invN - m * m;      // biased variance (torch BN fwd)
    mu[c]   = m;
    rstd[c] = rsqrtf(v + 1.0f);           // BN_EPS = 1.0
}

__global__ __launch_bounds__(256) void bn_apply_leaky(
    const float* __restrict__ x, const float* __restrict__ mu,
    const float* __restrict__ rstd, const float* __restrict__ gamma,
    const float* __restrict__ beta, float* __restrict__ y, long total)
{
    long i = (long)blockIdx.x * blockDim.x + threadIdx.x;
    if (i >= total) return;
    int c = (int)(i & (H - 1));
    float v = (x[i] - mu[c]) * rstd[c] * gamma[c] + beta[c];
    y[i] = (v >= 0.f) ? v : 0.01f * v;    // NEG_SLOPE
}

// ---------------------------------------------------------------------------
// Final FC: out[s,n,c] = x_s[n,:]@fcW[s,:,c] + fcb[s,c]
// ---------------------------------------------------------------------------
__global__ __launch_bounds__(256) void fc_out(
    const float* __restrict__ xA, const float* __restrict__ xB,
    const float* __restrict__ fcW, const float* __restrict__ fcb,
    float* __restrict__ out, int nN)
{
    long i = (long)blockIdx.x * blockDim.x + threadIdx.x;
    long total = 2L * nN * 16;
    if (i >= total) return;
    int c    = (int)(i & 15);
    long rem = i >> 4;
    int n    = (int)(rem % nN);
    int s    = (int)(rem / nN);
    const float* x = (s ? xB : xA) + (long)n * H;
    const float* w = fcW + (long)s * H * 16;
    float acc = fcb[s * 16 + c];
    for (int k = 0; k < H; ++k) acc += x[k] * w[k * 16 + c];
    out[i] = acc;
}

// ---------------------------------------------------------------------------
extern "C" void kernel_launch(void* const* d_in, const int* in_sizes, int n_in,
                              void* d_out, int out_size, void* d_ws, size_t ws_size,
                              hipStream_t stream)
{
    const float* xA_in  = (const float*)d_in[0];
    const float* xB_in  = (const float*)d_in[1];
    const float* Wsrc   = (const float*)d_in[2];
    const float* bsrc   = (const float*)d_in[3];
    const float* Wdst   = (const float*)d_in[4];
    const float* bdst   = (const float*)d_in[5];
    const float* Wupd   = (const float*)d_in[6];
    const float* bupd   = (const float*)d_in[7];
    const float* attnW1 = (const float*)d_in[8];
    const float* attnb1 = (const float*)d_in[9];
    const float* attnw2 = (const float*)d_in[10];
    const float* gamma  = (const float*)d_in[11];
    const float* beta   = (const float*)d_in[12];
    const float* fcW    = (const float*)d_in[13];
    const float* fcb    = (const float*)d_in[14];
    const int*   eAB    = (const int*)d_in[15];
    const int*   eBB    = (const int*)d_in[16];
    const int*   eBA    = (const int*)d_in[17];

    const int  nN = in_sizes[0] / H;        // 100000 (divisible by MTILE=32)
    const int  nE = in_sizes[15] / 2;       // 600000
    const long NH = (long)nN * H;

    // workspace layout (floats)
    float* ws    = (float*)d_ws;
    float* xA    = ws;
    float* xB    = xA + NH;
    float* embB1 = xB + NH;
    float* embB2 = embB1 + NH;
    float* embA  = embB2 + NH;
    float* aggr  = embA + NH;               // also reused as attn-combined xBnew
    float* cnt   = aggr + NH;               // nN floats
    float* Wd    = cnt + (((long)nN + 15) & ~15L);
    float* Wsf   = Wd + 6L * H * H;
    float* bf    = Wsf + 6L * H * H;
    float* gsum  = bf + 6 * H;
    float* gsq   = gsum + H;
    float* mu    = gsq + H;
    float* rstd  = mu + H;
    float* sums  = rstd + H;
    float* alpha = sums + 4;

    hipMemcpyAsync(xA, xA_in, NH * sizeof(float), hipMemcpyDeviceToDevice, stream);
    hipMemcpyAsync(xB, xB_in, NH * sizeof(float), hipMemcpyDeviceToDevice, stream);
    fold_weights<<<6, 256, 0, stream>>>(Wsrc, bsrc, Wdst, bdst, Wupd, bupd, Wd, Wsf, bf);

    const float invN = 1.0f / (float)nN;
    const long scatThreads = (long)nE * 32;
    const int  scatBlocks  = (int)((scatThreads + 255) / 256);
    const int  ewBlocks    = (int)((NH + 255) / 256);
    const int  gemmBlocks  = nN / MTILE;
    const int  bnBlocks    = (nN + BN_ROWS - 1) / BN_ROWS;

    auto conv = [&](const float* xsrc, const float* xdst, const int* e,
                    float* out, int lr) {
        hipMemsetAsync(aggr, 0, NH * sizeof(float), stream);
        hipMemsetAsync(cnt, 0, (size_t)nN * sizeof(float), stream);
        scatter_add<<<scatBlocks, 256, 0, stream>>>(xsrc, e, e + nE, aggr, cnt, nE);
        scatter_div<<<ewBlocks, 256, 0, stream>>>(aggr, cnt, NH);
        gemm_dual_bias<<<gemmBlocks, 256, 0, stream>>>(
            xdst, Wd + (long)lr * H * H, aggr, Wsf + (long)lr * H * H,
            bf + lr * H, out);
    };

    for (int l = 0; l < 2; ++l) {
        conv(xA, xB, eAB, embB1, l * 3 + 0);   // A -> B
        conv(xB, xB, eBB, embB2, l * 3 + 1);   // B -> B
        conv(xB, xA, eBA, embA,  l * 3 + 2);   // B -> A

        // attention combine over {embB1, embB2}
        hipMemsetAsync(sums, 0, 2 * sizeof(float), stream);
        long waves = 2L * nN;
        attn_score<<<(int)((waves * 32 + 255) / 256), 256, 0, stream>>>(
            embB1, embB2, attnW1 + (long)l * H * AA, attnb1 + l * AA,
            attnw2 + l * AA, sums, nN);
        attn_alpha<<<1, 1, 0, stream>>>(sums, alpha, invN);
        attn_combine<<<ewBlocks, 256, 0, stream>>>(embB1, embB2, alpha, aggr, NH);

        // BN + leaky for A
        hipMemsetAsync(gsum, 0, H * sizeof(float), stream);
        hipMemsetAsync(gsq, 0, H * sizeof(float), stream);
        bn_stats<<<bnBlocks, 256, 0, stream>>>(embA, gsum, gsq, nN);
        bn_finalize<<<1, H, 0, stream>>>(gsum, gsq, mu, rstd, invN);
        bn_apply_leaky<<<ewBlocks, 256, 0, stream>>>(
            embA, mu, rstd, gamma + (l * 2 + 0) * H, beta + (l * 2 + 0) * H, xA, NH);

        // BN + leaky for B
        hipMemsetAsync(gsum, 0, H * sizeof(float), stream);
        hipMemsetAsync(gsq, 0, H * sizeof(float), stream);
        bn_stats<<<bnBlocks, 256, 0, stream>>>(aggr, gsum, gsq, nN);
        bn_finalize<<<1, H, 0, stream>>>(gsum, gsq, mu, rstd, invN);
        bn_apply_leaky<<<ewBlocks, 256, 0, stream>>>(
            aggr, mu, rstd, gamma + (l * 2 + 1) * H, beta + (l * 2 + 1) * H, xB, NH);
    }

    long fcTotal = 2L * nN * 16;
    fc_out<<<(int)((fcTotal + 255) / 256), 256, 0, stream>>>(
        xA, xB, fcW, fcb, (float*)d_out, nN);
}